// Net_43937515438452
// MI455X (gfx1250) — compile-verified
//
#include <hip/hip_runtime.h>
#include <math.h>

// ---------------------------------------------------------------------------
// 50-layer LSTM, H=50, B=64, T=256, I=72 — layer-pipelined persistent kernel.
// fp32 WMMA (v_wmma_f32_16x16x4_f32). All GEMM operands are pre-packed into
// padded, fragment-order layouts so inner-loop loads are unconditional b64.
// Critical path: wavefront pipeline over (layer, t); ~306 stages instead of
// 12800 sequential cell steps. Elementwise uses v_exp_f32/v_rcp_f32 TRANS ops.
// ---------------------------------------------------------------------------

typedef float v2f __attribute__((ext_vector_type(2)));
typedef float v8f __attribute__((ext_vector_type(8)));

#define HDIM   50
#define G4H    200              // 4*H
#define NP     208              // N padded to 13 tiles
#define NL     50               // layers
#define BATCH  64
#define TSTEPS 256
#define IDIM   72
#define NTILES 13
#define KP     104              // [hprev(50)+pad2 | hin(50)+pad2]
#define KPH    52               // KP/2 (float2 rows)
#define EWITER 13               // ceil(64*50 / 256)
#define WSZ    (KPH * NP)       // packed weight float2s per layer = 10816

// fast sigmoid / tanh on the TRANS pipe (v_exp_f32 + v_rcp_f32);
// both saturate correctly for large |x| (exp -> inf -> rcp -> 0).
__device__ __forceinline__ float sigf(float x) {
  return __builtin_amdgcn_rcpf(1.0f + __expf(-x));
}
__device__ __forceinline__ float tanhfast(float x) {
  return 1.0f - 2.0f * __builtin_amdgcn_rcpf(__expf(2.0f * x) + 1.0f);
}

// ---------------------------------------------------------------------------
// Prep 1: pack per-layer recurrent weights into WMMA-B fragment order.
// Wpack[l][kp][n].x = B(2kp, n), .y = B(2kp+1, n); zero padded (k pads, n>=200).
// K layout: k in [0,50): w_hh[n][k]; [50,52): 0; [52,102): w_ih[n][k-52]; rest 0.
// Layer 0: hin segment is zero (input contribution comes via precomputed xg0).
// ---------------------------------------------------------------------------
__global__ __launch_bounds__(256) void pack_weights(const float* __restrict__ w_hh0,
                                                    const float* __restrict__ w_ih,
                                                    const float* __restrict__ w_hh,
                                                    const float* __restrict__ b_ih,
                                                    const float* __restrict__ b_hh,
                                                    v2f* __restrict__ Wpack,
                                                    float* __restrict__ biasp) {
  const int l = blockIdx.x;
  const float* wh = (l == 0) ? w_hh0 : (w_hh + (size_t)(l - 1) * G4H * HDIM);
  const float* wi = (l == 0) ? nullptr : (w_ih + (size_t)(l - 1) * G4H * HDIM);
  v2f* Wl = Wpack + (size_t)l * WSZ;
  for (int i = threadIdx.x; i < WSZ; i += 256) {
    const int kp = i / NP, n = i % NP;
    v2f v = {0.0f, 0.0f};
    if (n < G4H) {
      const int k0 = 2 * kp, k1 = 2 * kp + 1;
      if (k0 < HDIM) v.x = wh[(size_t)n * HDIM + k0];
      else if (k0 >= 52 && k0 < 52 + HDIM && wi) v.x = wi[(size_t)n * HDIM + (k0 - 52)];
      if (k1 < HDIM) v.y = wh[(size_t)n * HDIM + k1];
      else if (k1 >= 52 && k1 < 52 + HDIM && wi) v.y = wi[(size_t)n * HDIM + (k1 - 52)];
    }
    Wl[i] = v;
  }
  for (int n = threadIdx.x; n < NP; n += 256) {
    float b = 0.0f;
    if (l >= 1 && n < G4H) b = b_ih[(size_t)(l - 1) * G4H + n] + b_hh[(size_t)(l - 1) * G4H + n];
    biasp[l * NP + n] = b;
  }
}

// ---------------------------------------------------------------------------
// Prep 2: pack w_ih0 (for xg0 GEMM, K=72) + bias0 = b_ih0 + b_hh0, both padded.
// ---------------------------------------------------------------------------
__global__ __launch_bounds__(256) void pack_x(const float* __restrict__ w_ih0,
                                              const float* __restrict__ b_ih0,
                                              const float* __restrict__ b_hh0,
                                              v2f* __restrict__ WpackX,
                                              float* __restrict__ bias0p) {
  for (int i = threadIdx.x; i < 36 * NP; i += 256) {
    const int kp = i / NP, n = i % NP;
    v2f v = {0.0f, 0.0f};
    if (n < G4H) {
      v.x = w_ih0[(size_t)n * IDIM + 2 * kp];
      v.y = w_ih0[(size_t)n * IDIM + 2 * kp + 1];
    }
    WpackX[i] = v;
  }
  for (int n = threadIdx.x; n < NP; n += 256) {
    bias0p[n] = (n < G4H) ? (b_ih0[n] + b_hh0[n]) : 0.0f;
  }
}

// ---------------------------------------------------------------------------
// Kernel A: xg0p[t*64+b][0..207] = x @ w_ih0^T + bias0 (time-major, padded).
// M = 16384, N = 208, K = 72. One 16x16 tile per wave, all loads unconditional.
// ---------------------------------------------------------------------------
__global__ __launch_bounds__(256) void xg0_gemm(const float* __restrict__ x,
                                                const v2f* __restrict__ WpackX,
                                                const float* __restrict__ bias0p,
                                                float* __restrict__ xg0p) {
  const int tid   = threadIdx.x;
  const int lane  = tid & 31;
  const int gw    = blockIdx.x * 8 + (tid >> 5);   // 13312 waves total, exact
  const int mtile = gw / NTILES;
  const int ntile = gw % NTILES;
  const int lanem = lane & 15;
  const int kk    = (lane >> 4) * 2;
  const int mrow  = mtile * 16 + lanem;
  const int ncol  = ntile * 16 + lanem;

  const float bias = bias0p[ncol];
  v8f acc;
#pragma unroll
  for (int r = 0; r < 8; ++r) acc[r] = bias;

#pragma unroll
  for (int kc = 0; kc < IDIM; kc += 4) {
    v2f a = *(const v2f*)&x[(size_t)mrow * IDIM + kc + kk];   // b64, unconditional
    v2f b = WpackX[((kc + kk) >> 1) * NP + ncol];             // b64, unconditional
    acc = __builtin_amdgcn_wmma_f32_16x16x4_f32(false, a, false, b,
                                                (short)0, acc, false, false);
  }
  const int mbase = mtile * 16 + ((lane >> 4) << 3);          // C/D: vgpr r -> M=r / r+8
#pragma unroll
  for (int r = 0; r < 8; ++r) {
    const int rf = mbase + r;                 // flat row = b*256 + t
    const int bb = rf >> 8, tt = rf & 255;
    xg0p[(size_t)((tt << 6) + bb) * NP + ncol] = acc[r];
  }
}

// ---------------------------------------------------------------------------
// Pipelined persistent LSTM: one workgroup (one WGP) per layer, 8 wave32s.
// ---------------------------------------------------------------------------
__global__ __launch_bounds__(256, 1) void lstm_pipeline(
    const float* __restrict__ xg0p,
    const v2f*   __restrict__ Wpack,
    const float* __restrict__ biasp,
    float* __restrict__ buf0, float* __restrict__ buf1,
    int* __restrict__ flags) {
  __shared__ v2f   WldsV[WSZ];                      // 86528 B packed weights
  __shared__ float hprevL[BATCH][HDIM];             // 12800 B h_{t-1}
  __shared__ __align__(16) float hinL[BATCH][HDIM]; // 12800 B lower-layer h_t
  __shared__ float gL[BATCH][G4H];                  // 51200 B gate pre-activations
                                                    // total 163328 B LDS

  const int l    = blockIdx.x;
  const bool is0 = (l == 0);
  const int tid  = threadIdx.x;
  const int lane = tid & 31;
  const int wid  = tid >> 5;

  const v2f*   Wl  = Wpack + (size_t)l * WSZ;
  const float* bl  = biasp + l * NP;
  const float* bufin  = (l & 1) ? buf0 : buf1;      // layer l-1 wrote buf[(l-1)&1]
  float*       bufout = (l & 1) ? buf1 : buf0;

  const int mtile = wid & 3;
  const int nlo   = (wid < 4) ? 0 : 7;              // waves 0-3: ntiles 0..6; 4-7: 7..12
  const int lanem = lane & 15;
  const int kk    = (lane >> 4) * 2;                // 0 or 2
  const int mrow  = mtile * 16 + lanem;             // batch row for A frag
  const int mbase = mtile * 16 + ((lane >> 4) << 3);

  // stage packed weights to LDS (reused 256 steps); zero h0, hin, c0
  for (int i = tid; i < WSZ; i += 256) WldsV[i] = Wl[i];
  for (int i = tid; i < BATCH * HDIM; i += 256) {
    hprevL[i / HDIM][i % HDIM] = 0.0f;
    hinL[i / HDIM][i % HDIM]   = 0.0f;
  }
  float c_state[EWITER];
#pragma unroll
  for (int i = 0; i < EWITER; ++i) c_state[i] = 0.0f;

  // hoist loop-invariant bias into registers (layers >= 1)
  float biasReg[7];
#pragma unroll
  for (int nt = 0; nt < 7; ++nt)
    biasReg[nt] = (nlo + nt < NTILES) ? bl[(nlo + nt) * 16 + lanem] : 0.0f;

  __syncthreads();

  for (int t = 0; t < TSTEPS; ++t) {
    // ---- wait for lower layer's h_t, stage it into LDS (b128 copies) ----
    if (l > 0) {
      if (tid == 0) {
        while (__hip_atomic_load(&flags[l - 1], __ATOMIC_ACQUIRE,
                                 __HIP_MEMORY_SCOPE_AGENT) <= t) {
          __builtin_amdgcn_s_sleep(1);
        }
      }
      __syncthreads();
      const float4* src = (const float4*)(bufin + (size_t)t * BATCH * HDIM);
      float4* dst = (float4*)&hinL[0][0];
#pragma unroll
      for (int i = 0; i < 4; ++i) {                 // 800 float4 / 256 threads
        const int j = i * 256 + tid;
        if (j < (BATCH * HDIM) / 4) dst[j] = src[j];
      }
      __syncthreads();
    }

    // ---- acc init: bias regs (l>=1) or precomputed xg0 (l==0) ----
    v8f acc[7];
#pragma unroll
    for (int nt = 0; nt < 7; ++nt) {
      if (nlo + nt < NTILES) {
        if (is0) {
          const int n = (nlo + nt) * 16 + lanem;
#pragma unroll
          for (int r = 0; r < 8; ++r)
            acc[nt][r] = xg0p[(size_t)((t << 6) + mbase + r) * NP + n];
        } else {
          const float bb = biasReg[nt];
#pragma unroll
          for (int r = 0; r < 8; ++r) acc[nt][r] = bb;
        }
      }
    }

    // ---- fused GEMM: g = [h_prev ; h_in] @ Wpack, K=104, fully unrolled ----
#pragma unroll
    for (int kc = 0; kc < KP; kc += 4) {
      v2f a;
      if (kc < 48)       a = *(const v2f*)&hprevL[mrow][kc + kk];
      else if (kc == 48) { a = (v2f){0.0f, 0.0f}; if (kk == 0) a = *(const v2f*)&hprevL[mrow][48]; }
      else if (kc < 100) a = *(const v2f*)&hinL[mrow][kc + kk - 52];
      else               { a = (v2f){0.0f, 0.0f}; if (kk == 0) a = *(const v2f*)&hinL[mrow][48]; }
      const v2f* wrow = &WldsV[((kc + kk) >> 1) * NP];
#pragma unroll
      for (int nt = 0; nt < 7; ++nt) {
        if (nlo + nt < NTILES) {
          const int n = (nlo + nt) * 16 + lanem;
          acc[nt] = __builtin_amdgcn_wmma_f32_16x16x4_f32(false, a, false, wrow[n],
                                                          (short)0, acc[nt],
                                                          false, false);
        }
      }
    }

    // ---- scatter gate pre-activations to LDS (only n<200 real) ----
#pragma unroll
    for (int nt = 0; nt < 7; ++nt) {
      if (nlo + nt < NTILES) {
        const int n = (nlo + nt) * 16 + lanem;
        if (n < G4H) {
#pragma unroll
          for (int r = 0; r < 8; ++r) gL[mbase + r][n] = acc[nt][r];
        }
      }
    }
    __syncthreads();

    // ---- elementwise LSTM cell (c in registers, fixed (b,h) per thread) ----
#pragma unroll
    for (int it = 0; it < EWITER; ++it) {
      const int idx = it * 256 + tid;
      if (idx < BATCH * HDIM) {
        const int m = idx / HDIM, h = idx % HDIM;
        const float gi = gL[m][h];
        const float gf = gL[m][h + HDIM];
        const float gg = gL[m][h + 2 * HDIM];
        const float go = gL[m][h + 3 * HDIM];
        float c = sigf(gf) * c_state[it] + sigf(gi) * tanhfast(gg);
        c_state[it] = c;
        const float hv = sigf(go) * tanhfast(c);
        hprevL[m][h] = hv;
        bufout[(size_t)t * BATCH * HDIM + idx] = hv;   // coalesced
      }
    }
    __threadfence();   // publish h_t device-wide before flag release
    __syncthreads();
    if (tid == 0)
      __hip_atomic_store(&flags[l], t + 1, __ATOMIC_RELEASE,
                         __HIP_MEMORY_SCOPE_AGENT);
  }
}

// ---------------------------------------------------------------------------
// Final FC: out[64,2] = h_last @ fc_w^T + fc_b
// ---------------------------------------------------------------------------
__global__ void fc_kernel(const float* __restrict__ hlast,
                          const float* __restrict__ fc_w,
                          const float* __restrict__ fc_b,
                          float* __restrict__ out) {
  const int tid = threadIdx.x;         // 128 threads = 64 batches * 2 outputs
  const int b = tid >> 1, j = tid & 1;
  float s = fc_b[j];
  for (int k = 0; k < HDIM; ++k) s += hlast[b * HDIM + k] * fc_w[j * HDIM + k];
  out[b * 2 + j] = s;
}

__global__ void init_flags(int* __restrict__ flags) { flags[threadIdx.x] = 0; }

// ---------------------------------------------------------------------------
extern "C" void kernel_launch(void* const* d_in, const int* in_sizes, int n_in,
                              void* d_out, int out_size, void* d_ws, size_t ws_size,
                              hipStream_t stream) {
  const float* x     = (const float*)d_in[0];
  const float* w_ih0 = (const float*)d_in[1];
  const float* w_hh0 = (const float*)d_in[2];
  const float* b_ih0 = (const float*)d_in[3];
  const float* b_hh0 = (const float*)d_in[4];
  const float* w_ih  = (const float*)d_in[5];
  const float* w_hh  = (const float*)d_in[6];
  const float* b_ih  = (const float*)d_in[7];
  const float* b_hh  = (const float*)d_in[8];
  const float* fc_w  = (const float*)d_in[9];
  const float* fc_b  = (const float*)d_in[10];
  float* out = (float*)d_out;

  // workspace carve (floats): xg0p | buf0 | buf1 | biasp | bias0p | Wpack | WpackX | flags
  float* ws     = (float*)d_ws;
  float* xg0p   = ws;                                            // 3,407,872 f (16B aligned)
  float* buf0   = xg0p + (size_t)BATCH * TSTEPS * NP;            //   819,200 f
  float* buf1   = buf0 + (size_t)TSTEPS * BATCH * HDIM;          //   819,200 f
  float* biasp  = buf1 + (size_t)TSTEPS * BATCH * HDIM;          //    10,400 f
  float* bias0p = biasp + (size_t)NL * NP;                       //       208 f
  v2f*   Wpack  = (v2f*)(bias0p + NP);                           // 540,800 v2f (8B aligned)
  v2f*   WpackX = Wpack + (size_t)NL * WSZ;                      //   7,488 v2f
  int*   flags  = (int*)(WpackX + 36 * NP);

  init_flags<<<1, 64, 0, stream>>>(flags);
  pack_weights<<<NL, 256, 0, stream>>>(w_hh0, w_ih, w_hh, b_ih, b_hh, Wpack, biasp);
  pack_x<<<1, 256, 0, stream>>>(w_ih0, b_ih0, b_hh0, WpackX, bias0p);
  xg0_gemm<<<1664, 256, 0, stream>>>(x, WpackX, bias0p, xg0p);   // 13312 tiles
  lstm_pipeline<<<NL, 256, 0, stream>>>(xg0p, Wpack, biasp, buf0, buf1, flags);
  fc_kernel<<<1, 128, 0, stream>>>(buf1 + (size_t)(TSTEPS - 1) * BATCH * HDIM,
                                   fc_w, fc_b, out);
}